// BatchNormSPD_5712306504232
// MI455X (gfx1250) — compile-verified
//
#include <hip/hip_runtime.h>
#include <cstdint>

typedef __attribute__((ext_vector_type(2))) float v2f;
typedef __attribute__((ext_vector_type(8))) float v8f;

#define LDST 68   // padded LDS row stride (floats) to dodge 64-bank conflicts

// ---------------------------------------------------------------------------
// Stage 1: W = sqrtm(bias) @ isqrtm(running_mean) via Newton-Schulz (fp32)
// ---------------------------------------------------------------------------
__device__ __forceinline__ void mm64(const float* __restrict__ A,
                                     const float* __restrict__ Bm,
                                     float* __restrict__ C, int tid) {
  for (int i = tid; i < 4096; i += 1024) {
    const int r = i >> 6, cc = i & 63;
    const float* ar = &A[r << 6];
    float acc = 0.f;
#pragma unroll
    for (int k = 0; k < 64; ++k) acc = fmaf(ar[k], Bm[(k << 6) + cc], acc);
    C[i] = acc;
  }
}

__global__ __launch_bounds__(1024) void spd_prepare(const float* __restrict__ Rm,
                                                    const float* __restrict__ Bsp,
                                                    float* __restrict__ Wout) {
  __shared__ float Ybuf[4096], Zbuf[4096], Tbuf[4096], Y2buf[4096], Z2buf[4096];
  __shared__ float S1[4096];   // scaled isqrtm(running_mean)
  __shared__ float red[1024];
  const int tid = threadIdx.x;

  float c_bias = 0.f;
  for (int phase = 0; phase < 2; ++phase) {
    const float* src = (phase == 0) ? Rm : Bsp;
    // load + Frobenius norm
    float p = 0.f;
    for (int i = tid; i < 4096; i += 1024) { float v = src[i]; Ybuf[i] = v; p += v * v; }
    red[tid] = p;
    __syncthreads();
    for (int s = 512; s > 0; s >>= 1) { if (tid < s) red[tid] += red[tid + s]; __syncthreads(); }
    const float c = sqrtf(red[0]);
    const float inv = 1.f / c;
    __syncthreads();
    for (int i = tid; i < 4096; i += 1024) {
      Ybuf[i] *= inv;
      Zbuf[i] = ((i >> 6) == (i & 63)) ? 1.f : 0.f;
    }
    __syncthreads();
    for (int it = 0; it < 16; ++it) {
      mm64(Zbuf, Ybuf, Tbuf, tid);                       // T = Z*Y
      __syncthreads();
      for (int i = tid; i < 4096; i += 1024)
        Tbuf[i] = (((i >> 6) == (i & 63)) ? 1.5f : 0.f) - 0.5f * Tbuf[i];
      __syncthreads();
      mm64(Ybuf, Tbuf, Y2buf, tid);                      // Y = Y*T
      mm64(Tbuf, Zbuf, Z2buf, tid);                      // Z = T*Z
      __syncthreads();
      for (int i = tid; i < 4096; i += 1024) { Ybuf[i] = Y2buf[i]; Zbuf[i] = Z2buf[i]; }
      __syncthreads();
    }
    if (phase == 0) {
      const float s = rsqrtf(c);                         // isqrtm = Z / sqrt(c)
      for (int i = tid; i < 4096; i += 1024) S1[i] = Zbuf[i] * s;
      __syncthreads();
    } else {
      c_bias = sqrtf(c);                                 // sqrtm = Y * sqrt(c)
    }
  }
  // W = (c_bias * Ybuf) @ S1
  for (int i = tid; i < 4096; i += 1024) {
    const int r = i >> 6, cc = i & 63;
    const float* ar = &Ybuf[r << 6];
    float acc = 0.f;
#pragma unroll
    for (int k = 0; k < 64; ++k) acc = fmaf(ar[k], S1[(k << 6) + cc], acc);
    Wout[i] = acc * c_bias;
  }
}

// ---------------------------------------------------------------------------
// Stage 2: Y_b = W @ X_b @ W^T, batched, fp32 WMMA + async global->LDS
// ---------------------------------------------------------------------------
__device__ __forceinline__ uint32_t lds_addr(const void* p) {
  return (uint32_t)(uintptr_t)p;   // flat LDS addr low 32 bits == LDS offset
}

__device__ __forceinline__ void async_cp16(uint32_t dst, const void* src) {
  asm volatile("global_load_async_to_lds_b128 %0, %1, off"
               :: "v"(dst), "v"(src) : "memory");
}

__global__ __launch_bounds__(256, 4) void spd_apply(const float* __restrict__ Xg,
                                                    const float* __restrict__ Wg,
                                                    float* __restrict__ Yg,
                                                    int nmat) {
  __shared__ float Wl[64 * LDST];        // 17.0 KB
  __shared__ float Xl[4][64 * LDST];     // 68.0 KB
  __shared__ float Tl[8][16 * LDST];     // 34.0 KB

  const int tid  = threadIdx.x;
  const int wave = tid >> 5;
  const int lane = tid & 31;
  const int l16  = lane & 15;
  const int lh   = lane >> 4;
  const int mloc = wave >> 1;            // matrix within block: 0..3
  const int half = wave & 1;             // which 32-row half this wave owns
  const int gm   = blockIdx.x * 4 + mloc;

  // --- async-stage W (1024 x float4) ---
  for (int j = tid; j < 1024; j += 256) {
    const int row = j >> 4, c4 = (j & 15) << 2;
    async_cp16(lds_addr(&Wl[row * LDST + c4]), Wg + ((size_t)j << 2));
  }
  // --- async-stage 4 X matrices (4096 x float4, fully coalesced) ---
  {
    const size_t gbase = (size_t)blockIdx.x * 4 * 4096;
    for (int j = tid; j < 4096; j += 256) {
      const int mm = j >> 10;
      if (blockIdx.x * 4 + mm < nmat) {
        const int e = j & 1023;
        const int row = e >> 4, c4 = (e & 15) << 2;
        async_cp16(lds_addr(&Xl[mm][row * LDST + c4]), Xg + gbase + ((size_t)j << 2));
      }
    }
  }
  asm volatile("s_wait_asynccnt 0" ::: "memory");
  __syncthreads();

  if (gm >= nmat) return;                // wave-uniform guard
  const float* Xm = Xl[mloc];
  float* Tb = Tl[wave];
  float* Yo = Yg + (size_t)gm * 4096;

  for (int r = 0; r < 2; ++r) {
    const int mi = half * 2 + r;         // 16-row output block index (0..3)

    // ---- T[16mi:16mi+16, :] = W[16mi:,:] @ X ----
    v8f acc0 = {}, acc1 = {}, acc2 = {}, acc3 = {};
#pragma unroll 4
    for (int kk = 0; kk < 16; ++kk) {
      const int k0 = kk << 2;
      const float* ap = &Wl[(mi * 16 + l16) * LDST + k0 + 2 * lh];
      v2f a; a.x = ap[0]; a.y = ap[1];                     // A frag (row-major)
      const float* bp = &Xm[(k0 + 2 * lh) * LDST + l16];   // B frags (column)
      v2f b0; b0.x = bp[0];  b0.y = bp[LDST];
      v2f b1; b1.x = bp[16]; b1.y = bp[LDST + 16];
      v2f b2; b2.x = bp[32]; b2.y = bp[LDST + 32];
      v2f b3; b3.x = bp[48]; b3.y = bp[LDST + 48];
      acc0 = __builtin_amdgcn_wmma_f32_16x16x4_f32(false, a, false, b0, (short)0, acc0, false, false);
      acc1 = __builtin_amdgcn_wmma_f32_16x16x4_f32(false, a, false, b1, (short)0, acc1, false, false);
      acc2 = __builtin_amdgcn_wmma_f32_16x16x4_f32(false, a, false, b2, (short)0, acc2, false, false);
      acc3 = __builtin_amdgcn_wmma_f32_16x16x4_f32(false, a, false, b3, (short)0, acc3, false, false);
    }
    // spill T row-block into this wave's private LDS strip
#pragma unroll
    for (int v = 0; v < 8; ++v) {
      const int rr = (v + 8 * lh) * LDST + l16;
      Tb[rr +  0] = acc0[v];
      Tb[rr + 16] = acc1[v];
      Tb[rr + 32] = acc2[v];
      Tb[rr + 48] = acc3[v];
    }

    // ---- Y[16mi:16mi+16, :] = Trow @ W^T  (W^T B-frag == row-major read of W) ----
    v8f y0 = {}, y1 = {}, y2 = {}, y3 = {};
#pragma unroll 4
    for (int kk = 0; kk < 16; ++kk) {
      const int k0 = kk << 2;
      const float* ap = &Tb[l16 * LDST + k0 + 2 * lh];
      v2f a; a.x = ap[0]; a.y = ap[1];
      const float* b0p = &Wl[(l16      ) * LDST + k0 + 2 * lh];
      const float* b1p = &Wl[(l16 + 16 ) * LDST + k0 + 2 * lh];
      const float* b2p = &Wl[(l16 + 32 ) * LDST + k0 + 2 * lh];
      const float* b3p = &Wl[(l16 + 48 ) * LDST + k0 + 2 * lh];
      v2f b0; b0.x = b0p[0]; b0.y = b0p[1];
      v2f b1; b1.x = b1p[0]; b1.y = b1p[1];
      v2f b2; b2.x = b2p[0]; b2.y = b2p[1];
      v2f b3; b3.x = b3p[0]; b3.y = b3p[1];
      y0 = __builtin_amdgcn_wmma_f32_16x16x4_f32(false, a, false, b0, (short)0, y0, false, false);
      y1 = __builtin_amdgcn_wmma_f32_16x16x4_f32(false, a, false, b1, (short)0, y1, false, false);
      y2 = __builtin_amdgcn_wmma_f32_16x16x4_f32(false, a, false, b2, (short)0, y2, false, false);
      y3 = __builtin_amdgcn_wmma_f32_16x16x4_f32(false, a, false, b3, (short)0, y3, false, false);
    }
    // store Y row-block to global (row-major, streaming / non-temporal)
#pragma unroll
    for (int v = 0; v < 8; ++v) {
      const int grow = (mi * 16 + v + 8 * lh) * 64 + l16;
      __builtin_nontemporal_store(y0[v], &Yo[grow +  0]);
      __builtin_nontemporal_store(y1[v], &Yo[grow + 16]);
      __builtin_nontemporal_store(y2[v], &Yo[grow + 32]);
      __builtin_nontemporal_store(y3[v], &Yo[grow + 48]);
    }
  }
}

// ---------------------------------------------------------------------------
extern "C" void kernel_launch(void* const* d_in, const int* in_sizes, int n_in,
                              void* d_out, int out_size, void* d_ws, size_t ws_size,
                              hipStream_t stream) {
  const float* X  = (const float*)d_in[0];   // (B,64,64)
  const float* Rm = (const float*)d_in[1];   // (64,64)
  const float* Bs = (const float*)d_in[2];   // (64,64)
  float* Y = (float*)d_out;
  float* W = (float*)d_ws;                   // 4096 floats of scratch

  const int nmat = in_sizes[0] / 4096;

  spd_prepare<<<1, 1024, 0, stream>>>(Rm, Bs, W);

  const int nblocks = (nmat + 3) / 4;
  if (nblocks > 0)
    spd_apply<<<nblocks, 256, 0, stream>>>(X, W, Y, nmat);
}